// _BaseFilter_48859547959409
// MI455X (gfx1250) — compile-verified
//
#include <hip/hip_runtime.h>

typedef float v2f __attribute__((ext_vector_type(2)));
typedef float v8f __attribute__((ext_vector_type(8)));
typedef unsigned int v4u __attribute__((ext_vector_type(4)));
typedef int v8i __attribute__((ext_vector_type(8)));
typedef int v4i __attribute__((ext_vector_type(4)));

#define T_LEN   32768
#define PAD     15
#define N_EXT   (T_LEN + 2*PAD)      // 32798
#define NTHR    512                  // 16 wave32s
#define NWAVE   (NTHR/32)
#define CHUNK   65                   // 512*65 = 33280 >= N_EXT
#define NCHUNK  NTHR
#define NGROUP  (NCHUNK/16)          // 32
#define NTILE   ((CHUNK+15)/16)      // 5
#define NJOBS   (NGROUP*NTILE)       // 160 (exactly 10 per wave)
#define HPADLEN (NTILE*16)           // 80 (zero-padded past CHUNK)
#define NLVL    9                    // log2(512) scan levels

// LDS layout (float indices)
#define OFF_XE  0
#define SZ_XE   32800
#define OFF_H   (OFF_XE + SZ_XE)       // 80*4 homogeneous output responses
#define OFF_W   (OFF_H + HPADLEN*4)    // 512*4 chunk states (scan in place)
#define OFF_P   (OFF_W + NCHUNK*4)     // 9 * 4x4 row-major powers of A^CHUNK
#define OFF_RED (OFF_P + NLVL*16)      // 512 reduction scratch
#define LDS_FLT (OFF_RED + NTHR)
#define LDS_BYTES (LDS_FLT * 4)        // 143,296 B < 160 KB -> 2 blocks/WGP

struct Coef { float b0,b1,b2,b3,b4,a1,a2,a3,a4; };

// One lfilter pass (REV=false forward over xe, REV=true over reversed signal),
// fully in LDS, chunk-parallel with WMMA-based boundary correction.
template<bool REV>
__device__ __forceinline__ void iir_pass(float* __restrict__ s, int tid, const Coef& c)
{
    // ---- 1) zero-state recurrence over own chunk, in place (clamped trip count) ----
    float z1 = 0.f, z2 = 0.f, z3 = 0.f, z4 = 0.f;
    const int base = tid * CHUNK;
    int tmax = N_EXT - base;
    tmax = (tmax < 0) ? 0 : ((tmax > CHUNK) ? CHUNK : tmax);
    for (int t = 0; t < tmax; ++t) {
        const int r = base + t;
        const int pos = REV ? (N_EXT - 1 - r) : r;
        const float xv = s[OFF_XE + pos];
        const float y  = fmaf(c.b0, xv, z1);
        const float n1 = fmaf(-c.a1, y, fmaf(c.b1, xv, z2));
        const float n2 = fmaf(-c.a2, y, fmaf(c.b2, xv, z3));
        const float n3 = fmaf(-c.a3, y, fmaf(c.b3, xv, z4));
        const float n4 = fmaf(-c.a4, y, c.b4 * xv);
        s[OFF_XE + pos] = y;
        z1 = n1; z2 = n2; z3 = n3; z4 = n4;
    }
    s[OFF_W + 4*tid + 0] = z1;
    s[OFF_W + 4*tid + 1] = z2;
    s[OFF_W + 4*tid + 2] = z3;
    s[OFF_W + 4*tid + 3] = z4;
    __syncthreads();

    // ---- 2) Kogge-Stone scan of chunk states: w_i = (A^L) w_{i-1} + p_i ----
    for (int k = 0; k < NLVL; ++k) {
        const int o = 1 << k;
        float w0 = s[OFF_W + 4*tid + 0];
        float w1 = s[OFF_W + 4*tid + 1];
        float w2 = s[OFF_W + 4*tid + 2];
        float w3 = s[OFF_W + 4*tid + 3];
        if (tid >= o) {
            const float* P  = s + OFF_P + 16*k;
            const float* wp = s + OFF_W + 4*(tid - o);
            w0 += P[ 0]*wp[0] + P[ 1]*wp[1] + P[ 2]*wp[2] + P[ 3]*wp[3];
            w1 += P[ 4]*wp[0] + P[ 5]*wp[1] + P[ 6]*wp[2] + P[ 7]*wp[3];
            w2 += P[ 8]*wp[0] + P[ 9]*wp[1] + P[10]*wp[2] + P[11]*wp[3];
            w3 += P[12]*wp[0] + P[13]*wp[1] + P[14]*wp[2] + P[15]*wp[3];
        }
        __syncthreads();
        s[OFF_W + 4*tid + 0] = w0;
        s[OFF_W + 4*tid + 1] = w1;
        s[OFF_W + 4*tid + 2] = w2;
        s[OFF_W + 4*tid + 3] = w3;
        __syncthreads();
    }

    // ---- 3) boundary correction: y[t] += h[t] . z_in  via V_WMMA_F32_16X16X4_F32 ----
    // D(16 t's x 16 chunks) = H(16x4) * Z(4x16); z_in(chunk c) = w[c-1] (0 for c==0).
    const int lane = tid & 31;
    const int lo   = lane & 15;
    const int hi   = lane >> 4;
    const int k0   = hi << 1;        // K pair {0,1} for lanes 0-15, {2,3} for 16-31
    const int wave = tid >> 5;
    for (int job = wave; job < NJOBS; job += NWAVE) {
        const int g  = job / NTILE;
        const int tt = job - g * NTILE;
        const int t0 = tt * 16;
        const int c0 = g * 16;
        v2f A, B;
        const int ta = t0 + lo;                       // A row = time offset in chunk
        A.x = s[OFF_H + 4*ta + k0 + 0];
        A.y = s[OFF_H + 4*ta + k0 + 1];
        const int cb  = c0 + lo;                      // B col = chunk index
        const int ci  = (cb == 0) ? 0 : (cb - 1);     // clamped index (no exec churn)
        const float sel = (cb == 0) ? 0.f : 1.f;
        B.x = sel * s[OFF_W + 4*ci + k0 + 0];
        B.y = sel * s[OFF_W + 4*ci + k0 + 1];
        v8f Cacc = {0.f,0.f,0.f,0.f,0.f,0.f,0.f,0.f};
        v8f D = __builtin_amdgcn_wmma_f32_16x16x4_f32(
            false, A, false, B, (short)0, Cacc, false, false);

        const int cc   = c0 + lo;                     // D col N -> chunk
        const int tb   = t0 + 8*hi;                   // first time index this lane owns
        const int rg0  = cc * CHUNK + tb;
        // wave-uniform fast path: whole 8-element strip in range for every lane
        const bool full = (tt + 1 < NTILE) && ((c0 + 16) * CHUNK <= N_EXT);
        if (full) {
            const int pbase = REV ? (N_EXT - 1 - rg0) : rg0;
            #pragma unroll
            for (int rr = 0; rr < 8; ++rr) {
                const int pos = REV ? (pbase - rr) : (pbase + rr);
                s[OFF_XE + pos] += D[rr];
            }
        } else {
            #pragma unroll
            for (int rr = 0; rr < 8; ++rr) {
                const int t  = tb + rr;
                const int rg = rg0 + rr;
                if (t < CHUNK && rg < N_EXT) {
                    const int pos = REV ? (N_EXT - 1 - rg) : rg;
                    s[OFF_XE + pos] += D[rr];
                }
            }
        }
    }
    __syncthreads();
}

extern "C" __global__ __launch_bounds__(NTHR, 1)
void filtfilt_wmma(const float* __restrict__ x,
                   const float* __restrict__ bc,
                   const float* __restrict__ ac,
                   float* __restrict__ out)
{
    extern __shared__ float s[];
    const int tid = threadIdx.x;
    const long long row = blockIdx.x;
    const float* xr   = x   + row * (long long)T_LEN;
    float*       orow = out + row * (long long)T_LEN;

    const float inva0 = 1.f / ac[0];
    Coef c;
    c.b0 = bc[0]*inva0; c.b1 = bc[1]*inva0; c.b2 = bc[2]*inva0;
    c.b3 = bc[3]*inva0; c.b4 = bc[4]*inva0;
    c.a1 = ac[1]*inva0; c.a2 = ac[2]*inva0; c.a3 = ac[3]*inva0; c.a4 = ac[4]*inva0;

    // ---- Tensor Data Mover: DMA the whole row (32768 f32 = 128 KB) into LDS ----
    // D# per CDNA5 ISA: 2D tile 32768x1, dword elements.
    if (tid < 32) {   // wave 0 issues the DMA (TDM ignores EXEC; issue once per WG)
        const unsigned long long ga = (unsigned long long)(uintptr_t)xr;
        const unsigned int lds_base =
            (unsigned int)(uintptr_t)(s + OFF_XE + PAD);   // low 32 bits = LDS byte addr
        v4u g0;
        g0.x = 1u;                                          // count=1 (user descriptor)
        g0.y = lds_base;                                    // lds_addr [63:32]
        g0.z = (unsigned int)(ga & 0xffffffffu);            // global_addr low
        g0.w = ((unsigned int)(ga >> 32) & 0x01ffffffu)     // global_addr high [120:64]
             | 0x80000000u;                                 // type=2 ("image") [127:126]
        v8i g1;
        g1[0] = 0x00020000;          // workgroup_mask=0, data_size=2 (4 bytes)
        g1[1] = (int)0x80000000u;    // tensor_dim0[15:0]=0x8000 at [31:16]
        g1[2] = 0x00010000;          // tensor_dim0 hi=0; tensor_dim1[15:0]=1 at [31:16]
        g1[3] = (int)0x80000000u;    // tensor_dim1 hi=0; tile_dim0=0x8000 at [31:16]
        g1[4] = 1;                   // tile_dim1=1, tile_dim2=0
        g1[5] = 32768;               // tensor_dim0_stride low 32
        g1[6] = 0;                   // stride0 hi=0, tensor_dim1_stride low=0
        g1[7] = 0;                   // tensor_dim1_stride hi=0
        v4i gz4 = {0, 0, 0, 0};
        v8i gz8 = {0, 0, 0, 0, 0, 0, 0, 0};
        __builtin_amdgcn_tensor_load_to_lds(g0, g1, gz4, gz4, gz8, 0);
    }

    // ---- setup while the DMA runs (disjoint LDS regions) ----
    if (tid == 0) {
        // columns of per-sample transition A (DFII-T homogeneous part)
        float col[4][4] = {{-c.a1,-c.a2,-c.a3,-c.a4},
                           {1.f,0.f,0.f,0.f},
                           {0.f,1.f,0.f,0.f},
                           {0.f,0.f,1.f,0.f}};
        for (int st = 1; st < CHUNK; ++st) {
            #pragma unroll
            for (int j = 0; j < 4; ++j) {
                const float v0 = col[j][0], v1 = col[j][1], v2 = col[j][2], v3 = col[j][3];
                col[j][0] = fmaf(-c.a1, v0, v1);
                col[j][1] = fmaf(-c.a2, v0, v2);
                col[j][2] = fmaf(-c.a3, v0, v3);
                col[j][3] = -c.a4 * v0;
            }
        }
        float P[16];
        for (int r = 0; r < 4; ++r)
            for (int j = 0; j < 4; ++j)
                P[r*4 + j] = col[j][r];               // row-major A^CHUNK
        for (int i = 0; i < 16; ++i) s[OFF_P + i] = P[i];
        for (int k = 1; k < NLVL; ++k) {              // repeated squaring
            float Q[16];
            for (int r = 0; r < 4; ++r)
                for (int cc2 = 0; cc2 < 4; ++cc2) {
                    float acc = 0.f;
                    for (int m = 0; m < 4; ++m) acc = fmaf(P[r*4+m], P[m*4+cc2], acc);
                    Q[r*4 + cc2] = acc;
                }
            for (int i = 0; i < 16; ++i) { P[i] = Q[i]; s[OFF_P + 16*k + i] = Q[i]; }
        }
    }
    if (tid == 1) {   // homogeneous output response h[t] = e1^T A^t
        float h0 = 1.f, h1 = 0.f, h2 = 0.f, h3 = 0.f;
        for (int t = 0; t < HPADLEN; ++t) {
            if (t < CHUNK) {
                s[OFF_H + 4*t + 0] = h0; s[OFF_H + 4*t + 1] = h1;
                s[OFF_H + 4*t + 2] = h2; s[OFF_H + 4*t + 3] = h3;
                const float n0 = -(fmaf(c.a1, h0, fmaf(c.a2, h1, fmaf(c.a3, h2, c.a4*h3))));
                h3 = h2; h2 = h1; h1 = h0; h0 = n0;
            } else {
                s[OFF_H + 4*t + 0] = 0.f; s[OFF_H + 4*t + 1] = 0.f;
                s[OFF_H + 4*t + 2] = 0.f; s[OFF_H + 4*t + 3] = 0.f;
            }
        }
    }

    if (tid < 32) __builtin_amdgcn_s_wait_tensorcnt(0);   // wave 0: DMA done
    __syncthreads();                                      // LDS visible to all waves

    // ---- max|x| from LDS, reduce, normalize, odd extension ----
    float lmax = 0.f;
    #pragma unroll
    for (int j = 0; j < T_LEN/NTHR; ++j)
        lmax = fmaxf(lmax, fabsf(s[OFF_XE + PAD + tid + j*NTHR]));
    s[OFF_RED + tid] = lmax;
    __syncthreads();
    for (int st = NTHR/2; st > 0; st >>= 1) {
        if (tid < st) s[OFF_RED + tid] = fmaxf(s[OFF_RED + tid], s[OFF_RED + tid + st]);
        __syncthreads();
    }
    const float scale = s[OFF_RED];
    const float inv = 1.f / scale;
    for (int kk = tid; kk < T_LEN; kk += NTHR) s[OFF_XE + PAD + kk] *= inv;
    __syncthreads();
    if (tid < PAD) {
        // left:  xe[k] = 2*xn[0] - xn[15-k];  right: xe[PAD+T+k] = 2*xn[T-1] - xn[T-2-k]
        s[OFF_XE + tid] = fmaf(2.f, s[OFF_XE + PAD], -s[OFF_XE + 2*PAD - tid]);
        s[OFF_XE + PAD + T_LEN + tid] =
            fmaf(2.f, s[OFF_XE + PAD + T_LEN - 1], -s[OFF_XE + PAD + T_LEN - 2 - tid]);
    }
    __syncthreads();

    iir_pass<false>(s, tid, c);   // forward lfilter
    iir_pass<true >(s, tid, c);   // backward lfilter (result lands in natural order)

    // ---- write back (one HBM write per element), undo normalization ----
    float4* ov4 = (float4*)orow;
    #pragma unroll
    for (int j = 0; j < (T_LEN/4)/NTHR; ++j) {
        const int q = tid + j*NTHR;
        const int k = q * 4;
        float4 v;
        v.x = s[OFF_XE + PAD + k + 0] * scale;
        v.y = s[OFF_XE + PAD + k + 1] * scale;
        v.z = s[OFF_XE + PAD + k + 2] * scale;
        v.w = s[OFF_XE + PAD + k + 3] * scale;
        ov4[q] = v;
    }
}

extern "C" void kernel_launch(void* const* d_in, const int* in_sizes, int n_in,
                              void* d_out, int out_size, void* d_ws, size_t ws_size,
                              hipStream_t stream) {
    const float* x = (const float*)d_in[0];
    const float* b = (const float*)d_in[1];
    const float* a = (const float*)d_in[2];
    float* out = (float*)d_out;
    const int rows = in_sizes[0] / T_LEN;   // 32*16 = 512
    (void)n_in; (void)out_size; (void)d_ws; (void)ws_size;
    (void)hipFuncSetAttribute((const void*)filtfilt_wmma,
                              hipFuncAttributeMaxDynamicSharedMemorySize, LDS_BYTES);
    filtfilt_wmma<<<rows, NTHR, LDS_BYTES, stream>>>(x, b, a, out);
}